// SGStem_13778255086243
// MI455X (gfx1250) — compile-verified
//
#include <hip/hip_runtime.h>

// CDNA5 / gfx1250: wave32, f32 WMMA 16x16x4, async global->LDS gather pipeline.
typedef float v2f __attribute__((ext_vector_type(2)));
typedef float v8f __attribute__((ext_vector_type(8)));

#define HW      256
#define NPIX    (HW * HW)
#define FDIM    64
#define PIXB    16            // pixels per block = one WMMA M tile
#define CHUNK   128           // CSR entries per pipeline stage (== blockDim)
#define TOKS    80            // padded LDS row stride (floats): rows k and k+2
                              // hit disjoint bank groups for the two half-waves
#define NWAVES  4
#define BLOCKT  (NWAVES * 32)

__global__ __launch_bounds__(BLOCKT)
void sgstem_wmma_async(const int* __restrict__ indices,    // gene id per entry
                       const float* __restrict__ cnts,     // count per entry
                       const int* __restrict__ indptr,     // CSR row ptrs (NPIX+1)
                       const float* __restrict__ gamma,    // per-gene log-scale
                       const float* __restrict__ tokens,   // [N_GENES][FDIM]
                       float* __restrict__ out)            // [NPIX][FDIM]
{
    // double-buffered pipeline state
    __shared__ __align__(16) float tok_lds[2][CHUNK * TOKS]; // gathered token rows
    __shared__ __align__(16) float w_lds[2][CHUNK];          // cnt * exp(gamma)

    const int p0   = blockIdx.x * PIXB;
    const int t    = threadIdx.x;
    const int lane = t & 31;
    const int wave = t >> 5;
    const int m    = lane & 15;          // A row (pixel) / B col (feature)
    const int hi   = lane >> 4;          // half-wave selects the K pair
    const int col  = wave * 16 + m;      // feature column owned by this lane

    const int rowBeg = indptr[p0];
    const int rowEnd = indptr[p0 + PIXB];
    const int segBeg = indptr[p0 + m];
    const unsigned segLen = (unsigned)(indptr[p0 + m + 1] - segBeg);

    v8f acc = {0.f, 0.f, 0.f, 0.f, 0.f, 0.f, 0.f, 0.f};

    const int nCh = (rowEnd - rowBeg + CHUNK - 1) / CHUNK;

    if (nCh > 0) {
        // ---- prologue: metadata + async gather for chunk 0 ----
        {
            int gi = 0; float w = 0.f;
            const int e = rowBeg + t;
            if (e < rowEnd) { gi = indices[e]; w = cnts[e] * __expf(gamma[gi]); }
            w_lds[0][t] = w;
            const float*  gp = tokens + gi * FDIM;           // this thread's row
            const unsigned la = (unsigned)(size_t)&tok_lds[0][t * TOKS];
#pragma unroll
            for (int q = 0; q < 16; ++q)                     // 16 x B128 = 256 B
                asm volatile("global_load_async_to_lds_b128 %0, %1, off"
                             :: "v"(la + q * 16), "v"(gp + q * 4) : "memory");
        }

        for (int c = 0; c < nCh; ++c) {
            const int  buf      = c & 1;
            const int  basec    = rowBeg + c * CHUNK;
            const bool haveNext = (c + 1) < nCh;

            // next chunk's metadata into registers (overlaps with in-flight gather)
            int gi = 0; float wnext = 0.f;
            if (haveNext) {
                const int e = basec + CHUNK + t;
                if (e < rowEnd) { gi = indices[e]; wnext = cnts[e] * __expf(gamma[gi]); }
            }

            // gather(c) landed in LDS; all waves done reading the other buffer
            asm volatile("s_wait_asynccnt 0x0" ::: "memory");
            __syncthreads();

            // kick off gather(c+1) into the opposite buffer (no cross-thread deps)
            if (haveNext) {
                w_lds[1 - buf][t] = wnext;
                const float*  gp = tokens + gi * FDIM;
                const unsigned la = (unsigned)(size_t)&tok_lds[1 - buf][t * TOKS];
#pragma unroll
                for (int q = 0; q < 16; ++q)
                    asm volatile("global_load_async_to_lds_b128 %0, %1, off"
                                 :: "v"(la + q * 16), "v"(gp + q * 4) : "memory");
            }

            // ---- consume chunk c via V_WMMA_F32_16X16X4_F32 ----
            // A (16x4): lane L -> M = L&15, VGPR0 K=2*hi, VGPR1 K=2*hi+1.
            // A[m][k] = w[k] iff entry k is in pixel m's CSR segment; one
            // unsigned compare per element (no EXEC games around WMMA).
            const float* wbuf = w_lds[buf];
            const float* tbuf = tok_lds[buf];
            const int    cnt  = min(CHUNK, rowEnd - basec);
            const int    u0   = basec + 2 * hi - segBeg;
            for (int k0 = 0; k0 < cnt; k0 += 16) {
#pragma unroll
                for (int kk = 0; kk < 16; kk += 4) {
                    const int ka = k0 + kk + 2 * hi;         // even -> b64 load
                    const float2 wab = *(const float2*)(wbuf + ka);
                    const unsigned ua = (unsigned)(u0 + k0 + kk);
                    v2f a, b;
                    a.x = (ua     < segLen) ? wab.x : 0.f;
                    a.y = (ua + 1u < segLen) ? wab.y : 0.f;  // wrap-safe
                    b.x = tbuf[ka * TOKS + col];
                    b.y = tbuf[(ka + 1) * TOKS + col];
                    acc = __builtin_amdgcn_wmma_f32_16x16x4_f32(
                        /*neg_a=*/false, a, /*neg_b=*/false, b,
                        /*c_mod=*/(short)0, acc,
                        /*reuse_a=*/false, /*reuse_b=*/false);
                }
            }
        }
    }

    // ---- store D tile: VGPR r -> pixel p0 + r + 8*hi, feature wave*16 + m
    float* op = out + (p0 + 8 * hi) * FDIM + col;
#pragma unroll
    for (int r = 0; r < 8; ++r)
        op[r * FDIM] = acc[r];
}

extern "C" void kernel_launch(void* const* d_in, const int* in_sizes, int n_in,
                              void* d_out, int out_size, void* d_ws, size_t ws_size,
                              hipStream_t stream) {
    (void)in_sizes; (void)n_in; (void)out_size; (void)d_ws; (void)ws_size;
    const int*   indices = (const int*)  d_in[0];
    const float* cnts    = (const float*)d_in[1];
    const int*   indptr  = (const int*)  d_in[2];
    const float* gamma   = (const float*)d_in[3];
    const float* tokens  = (const float*)d_in[4];
    float*       out     = (float*)      d_out;

    sgstem_wmma_async<<<NPIX / PIXB, BLOCKT, 0, stream>>>(
        indices, cnts, indptr, gamma, tokens, out);
}